// Hypernetwork_PONG_dense_55181739819745
// MI455X (gfx1250) — compile-verified
//
#include <hip/hip_runtime.h>

typedef __attribute__((ext_vector_type(2))) float v2f;
typedef __attribute__((ext_vector_type(8))) float v8f;

#define D_IN   6400
#define NWAVES 8
#define KCHUNK (D_IN / NWAVES)   // 800 K-values per wave

// ---------------------------------------------------------------------------
// Stage A: x_scalar[row] = relu(dot(inputs[row,:], W1) + b1) for the 70 rows
// that actually influence the output. 5 tiles of 16 rows; tiles 0..3 cover
// rows 64*i (i=0..63), tile 4 covers rows (4096*m+1)/6 (m=0..5) for the w4
// branch. GEMV is done with V_WMMA_F32_16X16X4_F32: A = 16 rows x 4 K (f32),
// B = W1[k..k+3] broadcast across all 16 N columns, accumulate in f32.
// ---------------------------------------------------------------------------
__global__ __launch_bounds__(256) void pong_gemv_wmma(
    const float* __restrict__ inputs,   // (4096, 6400)
    const float* __restrict__ W1,       // (6400,)
    const float* __restrict__ b1,       // (1,)
    float* __restrict__ s_out)          // 80 floats: [0..63]=s3, [64..69]=s4
{
    const int tile = blockIdx.x;            // 0..4
    const int lane = threadIdx.x & 31;
    const int wid  = threadIdx.x >> 5;      // 0..7
    const int m16  = lane & 15;             // M index within tile
    const int half = lane >> 4;             // K sub-block (0: K0/K1, 1: K2/K3)

    // Batch row feeding this lane's A-matrix elements.
    int row;
    if (tile < 4) {
        row = 64 * (tile * 16 + m16);
    } else {
        row = (m16 < 6) ? (4096 * m16 + 1) / 6 : 0;   // {0,682,1365,2048,2730,3413}
    }

    const float* aptr = inputs + (size_t)row * D_IN + 2 * half;
    const float* bptr = W1 + 2 * half;

    v8f acc = {};
    const int kbeg = wid * KCHUNK;
    const int kend = kbeg + KCHUNK;
    for (int k = kbeg; k < kend; k += 4) {
        // A layout: lane<16 -> VGPR0=A[m,k+0], VGPR1=A[m,k+1];
        //           lane>=16 -> VGPR0=A[m,k+2], VGPR1=A[m,k+3]  (one b64 load)
        v2f a = *(const v2f*)(aptr + k);
        // B broadcast: every N column = W1, same b64 pattern per half-wave.
        v2f b = *(const v2f*)(bptr + k);
        acc = __builtin_amdgcn_wmma_f32_16x16x4_f32(
            /*neg_a=*/false, a, /*neg_b=*/false, b,
            /*c_mod=*/(short)0, acc, /*reuse_a=*/false, /*reuse_b=*/false);
    }

    // D layout: lane 0 holds D[0..7, N=0] in VGPRs 0..7, lane 16 holds D[8..15, N=0].
    __shared__ float red[NWAVES][16];
    if (lane == 0) {
        #pragma unroll
        for (int r = 0; r < 8; ++r) red[wid][r] = acc[r];
    }
    if (lane == 16) {
        #pragma unroll
        for (int r = 0; r < 8; ++r) red[wid][8 + r] = acc[r];
    }
    __syncthreads();

    if (threadIdx.x < 16) {
        float sum = 0.f;
        #pragma unroll
        for (int w = 0; w < NWAVES; ++w) sum += red[w][threadIdx.x];
        s_out[tile * 16 + threadIdx.x] = fmaxf(sum + b1[0], 0.f);
    }
}

// ---------------------------------------------------------------------------
// Stage B: expand 70 scalars into the 412742-element output.
//   out[0      .. 409599] = tile(sel3, 80), sel3[i*80+k] from s3[i]
//   out[409600 .. 409663] = 0
//   out[409664 .. 412741] = sel4[i*513+k] from s4[i]
// ---------------------------------------------------------------------------
__global__ __launch_bounds__(256) void pong_expand(
    const float* __restrict__ s,        // ws scalars
    const float* __restrict__ W2, const float* __restrict__ b2,
    const float* __restrict__ W3a, const float* __restrict__ b3a,
    const float* __restrict__ W3b, const float* __restrict__ b3b,
    const float* __restrict__ W4a, const float* __restrict__ b4a,
    const float* __restrict__ W4b, const float* __restrict__ b4b,
    float* __restrict__ out, int out_size)
{
    int idx = blockIdx.x * blockDim.x + threadIdx.x;
    if (idx >= out_size) return;

    float val;
    if (idx < 409600) {
        int j = idx % 5120;             // 80 identical tiles of sel3
        int i = j / 80;
        int k = j - i * 80;
        float a = fmaxf(s[i] * W2[1] + b2[1], 0.f);          // x[64i, 1]
        float h = fmaxf(a * W3a[0] + b3a[0], 0.f);
        val = fmaxf(h * W3b[k] + b3b[k], 0.f);
    } else if (idx < 409664) {
        val = 0.f;                      // bias = zeros(64)
    } else {
        int j = idx - 409664;
        int i = j / 513;
        int k = j - i * 513;
        int c = 64 + (4 * i + 1) % 6;   // column {65,69,67,65,69,67}
        float xv = fmaxf(s[64 + i] * W2[c] + b2[c], 0.f);
        float h  = fmaxf(xv * W4a[0] + b4a[0], 0.f);
        val = fmaxf(h * W4b[k] + b4b[k], 0.f);
    }
    out[idx] = val;
}

extern "C" void kernel_launch(void* const* d_in, const int* in_sizes, int n_in,
                              void* d_out, int out_size, void* d_ws, size_t ws_size,
                              hipStream_t stream) {
    const float* inputs = (const float*)d_in[0];
    const float* W1  = (const float*)d_in[1];
    const float* b1  = (const float*)d_in[2];
    const float* W2  = (const float*)d_in[3];
    const float* b2  = (const float*)d_in[4];
    const float* W3a = (const float*)d_in[5];
    const float* b3a = (const float*)d_in[6];
    const float* W3b = (const float*)d_in[7];
    const float* b3b = (const float*)d_in[8];
    const float* W4a = (const float*)d_in[9];
    const float* b4a = (const float*)d_in[10];
    const float* W4b = (const float*)d_in[11];
    const float* b4b = (const float*)d_in[12];
    float* out = (float*)d_out;
    float* s   = (float*)d_ws;          // 80 floats of scratch

    pong_gemv_wmma<<<5, 256, 0, stream>>>(inputs, W1, b1, s);

    int blocks = (out_size + 255) / 256;
    pong_expand<<<blocks, 256, 0, stream>>>(s, W2, b2, W3a, b3a, W3b, b3b,
                                            W4a, b4a, W4b, b4b, out, out_size);
}